// SorterishWrapper_29377576304967
// MI455X (gfx1250) — compile-verified
//
#include <hip/hip_runtime.h>
#include <hip/hip_bf16.h>

typedef float v2f __attribute__((ext_vector_type(2)));
typedef float v8f __attribute__((ext_vector_type(8)));

#define B_   512
#define C_   8
#define P_   24
#define F_   256
#define CF_  (C_ * F_)          // 2048
#define PERM_ELEMS (P_ * P_)    // 576

// ---------------------------------------------------------------------------
// Kernel A: per-batch scores -> soft_rank (argsort + PAV isotonic) -> perm
// ---------------------------------------------------------------------------
__global__ void __launch_bounds__(128)
score_rank_perm_kernel(const float* __restrict__ x_to_sort,
                       const float* __restrict__ W,
                       float* __restrict__ rank_out,
                       float* __restrict__ scores_out,
                       float* __restrict__ perm_ws) {
    const int b   = blockIdx.x;
    const int tid = threadIdx.x;      // 128 threads = 4 waves (wave32)

    __shared__ float Wl[CF_];
    __shared__ float sc[P_];
    __shared__ float rankL[P_];
    __shared__ float red[4];

    for (int i = tid; i < CF_; i += 128) Wl[i] = W[i];
    __syncthreads();

    const float* xb = x_to_sort + (size_t)b * C_ * P_ * F_;

    // ---- scores[p] = sum_{c,f} x[b,c,p,f] * W[c,f] ----
    for (int p = 0; p < P_; ++p) {
        float acc = 0.f;
        for (int i = tid; i < CF_; i += 128) {
            const int c = i >> 8;
            const int f = i & 255;
            acc += xb[(c * P_ + p) * F_ + f] * Wl[i];
        }
        // wave32 reduction
        for (int off = 16; off > 0; off >>= 1)
            acc += __shfl_xor(acc, off, 32);
        if ((tid & 31) == 0) red[tid >> 5] = acc;
        __syncthreads();
        if (tid == 0) sc[p] = red[0] + red[1] + red[2] + red[3];
        __syncthreads();
    }

    // ---- soft_rank on thread 0 (n = 24, serial) ----
    if (tid == 0) {
        float s[P_]; int pm[P_];
        for (int j = 0; j < P_; ++j) { s[j] = sc[j]; pm[j] = j; }
        // stable insertion sort, descending (== argsort(-z) stable)
        for (int i = 1; i < P_; ++i) {
            const float key = s[i]; const int kp = pm[i];
            int j = i - 1;
            while (j >= 0 && s[j] < key) { s[j+1] = s[j]; pm[j+1] = pm[j]; --j; }
            s[j+1] = key; pm[j+1] = kp;
        }
        // y = s - w,  w = [24, 23, ..., 1]
        float y[P_];
        for (int j = 0; j < P_; ++j) y[j] = s[j] - (float)(P_ - j);
        // PAV: non-increasing isotonic L2 regression (== min-max closed form)
        float bs[P_]; float bc[P_]; int nb = 0;
        for (int i = 0; i < P_; ++i) {
            bs[nb] = y[i]; bc[nb] = 1.f; ++nb;
            // merge while mean(prev) <= mean(cur)  (violation of non-increasing)
            while (nb > 1 && bs[nb-2] * bc[nb-1] <= bs[nb-1] * bc[nb-2]) {
                bs[nb-2] += bs[nb-1]; bc[nb-2] += bc[nb-1]; --nb;
            }
        }
        // primal = s - v ; scatter back through perm
        int idx = 0;
        for (int blk = 0; blk < nb; ++blk) {
            const float mean = bs[blk] / bc[blk];
            const int cnt = (int)bc[blk];
            for (int t = 0; t < cnt; ++t) {
                rankL[pm[idx]] = s[idx] - mean;
                ++idx;
            }
        }
        for (int j = 0; j < P_; ++j) {
            rank_out[b * P_ + j]   = rankL[j];
            scores_out[b * P_ + j] = sc[j];
        }
    }
    __syncthreads();

    // ---- perm[m,k] = exp(-(rank[k] - (m+1))^2) ----
    for (int e = tid; e < PERM_ELEMS; e += 128) {
        const int m = e / P_;
        const int k = e % P_;
        const float d = rankL[k] - (float)(m + 1);
        perm_ws[(size_t)b * PERM_ELEMS + e] = expf(-d * d);
    }
}

// ---------------------------------------------------------------------------
// Kernel B: sorted_x[b,c,m,f] = sum_k perm[b,m,k] * x[b,c,k,f]
// One wave per (b, c, 64-column group). V_WMMA_F32_16X16X4_F32, fp32 end-to-end.
// ---------------------------------------------------------------------------
__global__ void __launch_bounds__(256)
perm_gemm_kernel(const float* __restrict__ x_to_sort,
                 const float* __restrict__ x_not_sort,
                 const float* __restrict__ perm_ws,
                 float* __restrict__ out) {
    const int lane = threadIdx.x & 31;
    const int wid  = blockIdx.x * (blockDim.x >> 5) + (threadIdx.x >> 5);
    // wid in [0, 512*16*4)
    const int ng = wid & 3;            // 64-column group
    const int c  = (wid >> 2) & 15;    // concat channel
    const int b  = wid >> 6;           // batch
    const int f0 = ng * 64;

    const float* X = (c < 8)
        ? x_to_sort  + (size_t)(b * C_ + c)       * P_ * F_
        : x_not_sort + (size_t)(b * C_ + (c - 8)) * P_ * F_;
    const float* Pm = perm_ws + (size_t)b * PERM_ELEMS;

    const int mrow  = lane & 15;          // M (A / row of D) or N (B / col of D)
    const int khalf = (lane >> 4) * 2;    // K pair base within k-step

    // A fragments (perm): [mtile][kstep], ISA 16x4 fp32 layout
    v2f A[2][6];
    for (int mt = 0; mt < 2; ++mt) {
        const int m = mt * 16 + mrow;
        for (int ks = 0; ks < 6; ++ks) {
            const int k = ks * 4 + khalf;
            v2f a;
            a.x = (m < P_) ? Pm[m * P_ + k]     : 0.f;
            a.y = (m < P_) ? Pm[m * P_ + k + 1] : 0.f;
            A[mt][ks] = a;
        }
    }

    const v8f vzero = {0.f, 0.f, 0.f, 0.f, 0.f, 0.f, 0.f, 0.f};
    v8f acc[2][4];
    for (int mt = 0; mt < 2; ++mt)
        for (int nt = 0; nt < 4; ++nt) acc[mt][nt] = vzero;

    // K loop: K = 24 = 6 steps of 4; X streamed from HBM exactly once
    for (int ks = 0; ks < 6; ++ks) {
        const int k = ks * 4 + khalf;
        v2f Bf[4];
        for (int nt = 0; nt < 4; ++nt) {
            const int f = f0 + nt * 16 + mrow;     // N = lane%16
            v2f t;
            t.x = X[k * F_ + f];          // VGPR0: K = khalf
            t.y = X[(k + 1) * F_ + f];    // VGPR1: K = khalf + 1
            Bf[nt] = t;
        }
        for (int mt = 0; mt < 2; ++mt)
            for (int nt = 0; nt < 4; ++nt)
                acc[mt][nt] = __builtin_amdgcn_wmma_f32_16x16x4_f32(
                    /*neg_a=*/false, A[mt][ks],
                    /*neg_b=*/false, Bf[nt],
                    /*c_mod=*/(short)0, acc[mt][nt],
                    /*reuse_a=*/false, /*reuse_b=*/false);
    }

    // Store D: VGPR j, lane L -> M = mt*16 + j + (L/16)*8, N = f0+nt*16+L%16
    const int rowadd = (lane >> 4) * 8;
    float* outbc = out + (size_t)(b * 16 + c) * P_ * F_;
    for (int mt = 0; mt < 2; ++mt) {
        for (int nt = 0; nt < 4; ++nt) {
            const int f = f0 + nt * 16 + mrow;
            #pragma unroll
            for (int j = 0; j < 8; ++j) {
                const int M = mt * 16 + j + rowadd;
                if (M < P_) outbc[M * F_ + f] = acc[mt][nt][j];
            }
        }
    }
}

// ---------------------------------------------------------------------------
extern "C" void kernel_launch(void* const* d_in, const int* in_sizes, int n_in,
                              void* d_out, int out_size, void* d_ws, size_t ws_size,
                              hipStream_t stream) {
    const float* x_not_sort = (const float*)d_in[0];
    const float* x_to_sort  = (const float*)d_in[1];
    const float* W_score    = (const float*)d_in[2];

    float* out        = (float*)d_out;
    const size_t sorted_elems = (size_t)B_ * (2 * C_) * P_ * F_;  // 50331648
    float* rank_out   = out + sorted_elems;
    float* scores_out = rank_out + (size_t)B_ * P_;

    float* perm_ws = (float*)d_ws;    // 512*576 floats = 1.18 MB

    // Phase 1: scores -> soft_rank -> perm matrix
    score_rank_perm_kernel<<<B_, 128, 0, stream>>>(
        x_to_sort, W_score, rank_out, scores_out, perm_ws);

    // Phase 2: batched (24x24)x(24x256) GEMMs via fp32 WMMA
    const int total_waves = B_ * 16 * 4;           // 32768
    const int blocks = total_waves / 8;            // 8 waves per 256-thread block
    perm_gemm_kernel<<<blocks, 256, 0, stream>>>(
        x_to_sort, x_not_sort, perm_ws, out);
}